// SAGE_model_81200651698325
// MI455X (gfx1250) — compile-verified
//
#include <hip/hip_runtime.h>
#include <math.h>

// ---------------------------------------------------------------------------
// GraphSAGE (2x SAGEConv mean-aggr + linear head) for MI455X / gfx1250.
//
// Bottleneck is the edge gather/scatter (memory + f32-atomic bound); GEMMs
// (~10 GFLOP) run on V_WMMA_F32_16X16X4_F32 (fp32 matches reference).
// Layer-2 exploits linearity: mean(h1[src]) @ W2l == mean((h1@W2l)[src]),
// so we project 128->64 BEFORE scattering: halves layer-2 gather bytes and
// atomic count.
// ---------------------------------------------------------------------------

typedef __attribute__((ext_vector_type(2))) float v2f;
typedef __attribute__((ext_vector_type(8))) float v8f;

__device__ __forceinline__ v8f wmma_f32_16x16x4(v2f a, v2f b, v8f c) {
  // 8 args: (neg_a, A, neg_b, B, c_mod, C, reuse_a, reuse_b)
  return __builtin_amdgcn_wmma_f32_16x16x4_f32(false, a, false, b, (short)0, c,
                                               false, false);
}

// ---------------------------------------------------------------------------
__global__ void zero_f32_kernel(float* __restrict__ p, int n) {
  int i = blockIdx.x * blockDim.x + threadIdx.x;
  if (i < n) p[i] = 0.0f;
}

// Count in-degree per dst node (as float, accumulated with HW atomics).
__global__ void count_kernel(const int* __restrict__ dst,
                             float* __restrict__ cnt, int n_edges) {
  int e = blockIdx.x * blockDim.x + threadIdx.x;
  if (e < n_edges) unsafeAtomicAdd(&cnt[dst[e]], 1.0f);
}

// cnt -> 1/max(cnt,1) in place.
__global__ void inv_kernel(float* __restrict__ cnt, int n) {
  int i = blockIdx.x * blockDim.x + threadIdx.x;
  if (i < n) cnt[i] = 1.0f / fmaxf(cnt[i], 1.0f);
}

// Scatter-add: agg[dst[e]][:] += feat[src[e]][:]   (DIM = 128)
// One wave per edge; each lane moves a float4 (coalesced 512B per wave row).
__global__ void scatter128_kernel(const float* __restrict__ feat,
                                  const int* __restrict__ src,
                                  const int* __restrict__ dst,
                                  float* __restrict__ agg, int n_edges) {
  int tid = blockIdx.x * blockDim.x + threadIdx.x;
  int e = tid >> 5;
  if (e >= n_edges) return;
  int lane = tid & 31;
  size_t srow = (size_t)src[e] * 128 + lane * 4;
  size_t drow = (size_t)dst[e] * 128 + lane * 4;
  float4 v = *(const float4*)(feat + srow);
  unsafeAtomicAdd(&agg[drow + 0], v.x);
  unsafeAtomicAdd(&agg[drow + 1], v.y);
  unsafeAtomicAdd(&agg[drow + 2], v.z);
  unsafeAtomicAdd(&agg[drow + 3], v.w);
}

// Scatter-add for 64-dim features: one wave per edge, float2 per lane.
__global__ void scatter64_kernel(const float* __restrict__ feat,
                                 const int* __restrict__ src,
                                 const int* __restrict__ dst,
                                 float* __restrict__ agg, int n_edges) {
  int tid = blockIdx.x * blockDim.x + threadIdx.x;
  int e = tid >> 5;
  if (e >= n_edges) return;
  int lane = tid & 31;
  size_t srow = (size_t)src[e] * 64 + lane * 2;
  size_t drow = (size_t)dst[e] * 64 + lane * 2;
  float2 v = *(const float2*)(feat + srow);
  unsafeAtomicAdd(&agg[drow + 0], v.x);
  unsafeAtomicAdd(&agg[drow + 1], v.y);
}

// ---------------------------------------------------------------------------
// Layer 1: out[m][n] = relu( sum_k (agg[m][k]*inv[m]) * Wl[k][n]
//                          + sum_k  xin[m][k]         * Wr[k][n] + bias[n] )
// K = 128, NOUT = 128. One wave per 16x16 tile; block = 16-row strip.
// ---------------------------------------------------------------------------
template <int NOUT>
__global__ void sage_wmma_kernel(const float* __restrict__ agg,
                                 const float* __restrict__ inv,
                                 const float* __restrict__ xin,
                                 const float* __restrict__ Wl,
                                 const float* __restrict__ bias,
                                 const float* __restrict__ Wr,
                                 float* __restrict__ out, int n_nodes) {
  const int lane = threadIdx.x & 31;
  const int wave = threadIdx.x >> 5;
  const int row0 = blockIdx.x * 16;
  const int col0 = wave * 16;

  const int m = row0 + (lane & 15);
  const int mc = m < n_nodes ? m : n_nodes - 1;  // clamp loads, EXEC stays full
  const int n = col0 + (lane & 15);
  const int khalf = (lane >> 4) * 2;

  const float invm = inv[mc];
  const float* arow = agg + (size_t)mc * 128;
  const float* xrow = xin + (size_t)mc * 128;

  v8f acc = {0.f, 0.f, 0.f, 0.f, 0.f, 0.f, 0.f, 0.f};

#pragma unroll 8
  for (int k0 = 0; k0 < 128; k0 += 4) {
    int kb = k0 + khalf;
    v2f a = *(const v2f*)(arow + kb);
    a *= invm;
    v2f b;
    b.x = Wl[(size_t)kb * NOUT + n];
    b.y = Wl[(size_t)(kb + 1) * NOUT + n];
    acc = wmma_f32_16x16x4(a, b, acc);
  }
#pragma unroll 8
  for (int k0 = 0; k0 < 128; k0 += 4) {
    int kb = k0 + khalf;
    v2f a = *(const v2f*)(xrow + kb);
    v2f b;
    b.x = Wr[(size_t)kb * NOUT + n];
    b.y = Wr[(size_t)(kb + 1) * NOUT + n];
    acc = wmma_f32_16x16x4(a, b, acc);
  }

  const float bn = bias[n];
  const int mbase = row0 + (lane >> 4) * 8;
#pragma unroll
  for (int r = 0; r < 8; ++r) {
    int mm = mbase + r;
    if (mm < n_nodes)
      out[(size_t)mm * NOUT + n] = fmaxf(acc[r] + bn, 0.0f);
  }
}

// Plain GEMM: out = Xin[nx128] @ W[128xNOUT] (no bias / activation).
template <int NOUT>
__global__ void gemm_wmma_kernel(const float* __restrict__ xin,
                                 const float* __restrict__ W,
                                 float* __restrict__ out, int n_nodes) {
  const int lane = threadIdx.x & 31;
  const int wave = threadIdx.x >> 5;
  const int row0 = blockIdx.x * 16;
  const int col0 = wave * 16;

  const int m = row0 + (lane & 15);
  const int mc = m < n_nodes ? m : n_nodes - 1;
  const int n = col0 + (lane & 15);
  const int khalf = (lane >> 4) * 2;
  const float* xrow = xin + (size_t)mc * 128;

  v8f acc = {0.f, 0.f, 0.f, 0.f, 0.f, 0.f, 0.f, 0.f};
#pragma unroll 8
  for (int k0 = 0; k0 < 128; k0 += 4) {
    int kb = k0 + khalf;
    v2f a = *(const v2f*)(xrow + kb);
    v2f b;
    b.x = W[(size_t)kb * NOUT + n];
    b.y = W[(size_t)(kb + 1) * NOUT + n];
    acc = wmma_f32_16x16x4(a, b, acc);
  }

  const int mbase = row0 + (lane >> 4) * 8;
#pragma unroll
  for (int r = 0; r < 8; ++r) {
    int mm = mbase + r;
    if (mm < n_nodes) out[(size_t)mm * NOUT + n] = acc[r];
  }
}

// Layer 2 combine: h2[m][n] = relu( agg64[m][n]*inv[m]
//                                 + sum_k h1[m][k]*W2r[k][n] + b2[n] )
// (agg64 already holds segsum of h1@W2l projected features.)  NOUT = 64.
__global__ void sage2_wmma_kernel(const float* __restrict__ agg64,
                                  const float* __restrict__ inv,
                                  const float* __restrict__ h1,
                                  const float* __restrict__ W2r,
                                  const float* __restrict__ bias,
                                  float* __restrict__ out, int n_nodes) {
  constexpr int NOUT = 64;
  const int lane = threadIdx.x & 31;
  const int wave = threadIdx.x >> 5;
  const int row0 = blockIdx.x * 16;
  const int col0 = wave * 16;

  const int m = row0 + (lane & 15);
  const int mc = m < n_nodes ? m : n_nodes - 1;
  const int n = col0 + (lane & 15);
  const int khalf = (lane >> 4) * 2;
  const float* xrow = h1 + (size_t)mc * 128;

  v8f acc = {0.f, 0.f, 0.f, 0.f, 0.f, 0.f, 0.f, 0.f};
#pragma unroll 8
  for (int k0 = 0; k0 < 128; k0 += 4) {
    int kb = k0 + khalf;
    v2f a = *(const v2f*)(xrow + kb);
    v2f b;
    b.x = W2r[(size_t)kb * NOUT + n];
    b.y = W2r[(size_t)(kb + 1) * NOUT + n];
    acc = wmma_f32_16x16x4(a, b, acc);
  }

  const float bn = bias[n];
  const int mbase = row0 + (lane >> 4) * 8;
#pragma unroll
  for (int r = 0; r < 8; ++r) {
    int mm = mbase + r;
    if (mm < n_nodes) {
      float mean = agg64[(size_t)mm * NOUT + n] * inv[mm];
      out[(size_t)mm * NOUT + n] = fmaxf(acc[r] + mean + bn, 0.0f);
    }
  }
}

// out[i] = sigmoid( dot(h2[i,0:64], Wfc) + bfc )
__global__ void head_kernel(const float* __restrict__ h2,
                            const float* __restrict__ Wfc,
                            const float* __restrict__ bfc,
                            float* __restrict__ out, int n_nodes) {
  int i = blockIdx.x * blockDim.x + threadIdx.x;
  if (i >= n_nodes) return;
  const float* row = h2 + (size_t)i * 64;
  float s = bfc[0];
#pragma unroll
  for (int j = 0; j < 64; j += 4) {
    float4 v = *(const float4*)(row + j);
    float4 w = *(const float4*)(Wfc + j);
    s += v.x * w.x + v.y * w.y + v.z * w.z + v.w * w.w;
  }
  out[i] = 1.0f / (1.0f + __expf(-s));
}

// ---------------------------------------------------------------------------
static inline size_t align256(size_t x) { return (x + 255) & ~(size_t)255; }

extern "C" void kernel_launch(void* const* d_in, const int* in_sizes, int n_in,
                              void* d_out, int out_size, void* d_ws,
                              size_t ws_size, hipStream_t stream) {
  const float* x   = (const float*)d_in[0];
  const int*   ei  = (const int*)d_in[1];
  const float* W1l = (const float*)d_in[2];
  const float* b1  = (const float*)d_in[3];
  const float* W1r = (const float*)d_in[4];
  const float* W2l = (const float*)d_in[5];
  const float* b2  = (const float*)d_in[6];
  const float* W2r = (const float*)d_in[7];
  const float* Wfc = (const float*)d_in[8];
  const float* bfc = (const float*)d_in[9];

  const int n_nodes = in_sizes[0] / 128;
  const int n_edges = in_sizes[1] / 2;
  const int* src = ei;             // edge_index[0]
  const int* dst = ei + n_edges;   // edge_index[1]

  // Workspace layout
  char* ws = (char*)d_ws;
  size_t off = 0;
  float* inv = (float*)(ws + off); off += align256((size_t)n_nodes * 4);
  float* agg = (float*)(ws + off); off += align256((size_t)n_nodes * 128 * 4);
  float* h1  = (float*)(ws + off); off += align256((size_t)n_nodes * 128 * 4);
  float* h2  = (float*)(ws + off); off += align256((size_t)n_nodes * 64 * 4);
  float* p   = (float*)(ws + off); off += align256((size_t)n_nodes * 64 * 4);
  (void)ws_size;

  const int TPB = 256;
  const int scatterBlocks = (n_edges * 32 + TPB - 1) / TPB;
  const int strips = (n_nodes + 15) / 16;

  // Degree counts -> reciprocal (shared by both layers)
  zero_f32_kernel<<<(n_nodes + TPB - 1) / TPB, TPB, 0, stream>>>(inv, n_nodes);
  count_kernel<<<(n_edges + TPB - 1) / TPB, TPB, 0, stream>>>(dst, inv, n_edges);
  inv_kernel<<<(n_nodes + TPB - 1) / TPB, TPB, 0, stream>>>(inv, n_nodes);

  // Layer 1: agg = segsum(x[src] -> dst); h1 = relu(mean@W1l + b1 + x@W1r)
  {
    int nAgg = n_nodes * 128;
    zero_f32_kernel<<<(nAgg + TPB - 1) / TPB, TPB, 0, stream>>>(agg, nAgg);
    scatter128_kernel<<<scatterBlocks, TPB, 0, stream>>>(x, src, dst, agg,
                                                         n_edges);
    sage_wmma_kernel<128><<<strips, 256, 0, stream>>>(agg, inv, x, W1l, b1,
                                                      W1r, h1, n_nodes);
  }

  // Layer 2 (projected-first): p = h1 @ W2l (64-dim); agg64 = segsum(p[src]);
  // h2 = relu(agg64*inv + h1@W2r + b2)
  {
    gemm_wmma_kernel<64><<<strips, 128, 0, stream>>>(h1, W2l, p, n_nodes);
    int nAgg = n_nodes * 64;
    zero_f32_kernel<<<(nAgg + TPB - 1) / TPB, TPB, 0, stream>>>(agg, nAgg);
    scatter64_kernel<<<scatterBlocks, TPB, 0, stream>>>(p, src, dst, agg,
                                                        n_edges);
    sage2_wmma_kernel<<<strips, 128, 0, stream>>>(agg, inv, h1, W2r, b2, h2,
                                                  n_nodes);
  }

  // Head: out = sigmoid(h2 @ Wfc + bfc)
  head_kernel<<<(n_nodes + TPB - 1) / TPB, TPB, 0, stream>>>(h2, Wfc, bfc,
                                                             (float*)d_out,
                                                             n_nodes);
}